// MPSSSMLayer_8306466750955
// MI455X (gfx1250) — compile-verified
//
#include <hip/hip_runtime.h>
#include <hip/hip_bf16.h>
#include <math.h>

#define BATCH 8
#define SEQ   4096
#define D_IN  32
#define D_STATE 16
#define DT_RANK 16
#define HID   128
#define ROWS  (BATCH*SEQ)            /* 32768 flattened (b,s) rows */
#define OUT_MAIN ((size_t)ROWS*D_IN*D_IN) /* 33554432 */

typedef float v2f __attribute__((ext_vector_type(2)));
typedef float v8f __attribute__((ext_vector_type(8)));

static __device__ __forceinline__ v8f wmma4(v2f a, v2f b, v8f c) {
  // D = A(16x4 f32) * B(4x16 f32) + C(16x16 f32) -> v_wmma_f32_16x16x4_f32
  return __builtin_amdgcn_wmma_f32_16x16x4_f32(
      /*neg_a=*/false, a, /*neg_b=*/false, b,
      /*c_mod=*/(short)0, c, /*reuse_a=*/false, /*reuse_b=*/false);
}

// ---------------------------------------------------------------------------
// Kernel 1: per 16-row tile, compute
//   xp = x@Win + bin          (two 16x16 WMMA accumulators, K=32)
//   dtlin = xp@Wdt, Bm = xp@WB, Cm = xp@WC   (shared A-fragments, K=32, N=16)
//   dt = softplus((dtlin+bdt)@Wdtr) ; Ad = exp(dt*A) ; Bd = dt*Bm
// One wave32 per block. WMMA requires EXEC all ones -> full 32-thread block.
// ---------------------------------------------------------------------------
__global__ void __launch_bounds__(32)
proj_kernel(const float* __restrict__ x, const float* __restrict__ Avec,
            const float* __restrict__ Win, const float* __restrict__ bin_,
            const float* __restrict__ Wdt, const float* __restrict__ bdt,
            const float* __restrict__ Wdtr,
            const float* __restrict__ WB, const float* __restrict__ WC,
            float* __restrict__ xp_out, float* __restrict__ Ad_out,
            float* __restrict__ Bd_out, float* __restrict__ C_out)
{
  __shared__ float sxp[16][33];
  __shared__ float sdt[16][17];
  __shared__ float sB[16][17];
  __shared__ float sC[16][17];
  __shared__ float sdtv[16];

  const int lane = threadIdx.x;
  const int lo = lane & 15;        // A: M row / C,D: N col
  const int hi = lane >> 4;        // half-wave selects K pair (A/B), M+8 (C/D)
  const int row0 = blockIdx.x * 16;

  // ---- Stage 1: xp = x @ Win + bin  (N tiles {0..15},{16..31})
  v8f acc0, acc1;
  const float b0v = bin_[lo], b1v = bin_[16 + lo];
  #pragma unroll
  for (int v = 0; v < 8; ++v) { acc0[v] = b0v; acc1[v] = b1v; }

  #pragma unroll
  for (int k0 = 0; k0 < D_IN; k0 += 4) {
    const int k = k0 + 2 * hi;
    v2f a;  a.x  = x[(size_t)(row0 + lo) * D_IN + k];
            a.y  = x[(size_t)(row0 + lo) * D_IN + k + 1];
    v2f bA; bA.x = Win[k * D_IN + lo];      bA.y = Win[(k + 1) * D_IN + lo];
    v2f bB; bB.x = Win[k * D_IN + 16 + lo]; bB.y = Win[(k + 1) * D_IN + 16 + lo];
    acc0 = wmma4(a, bA, acc0);
    acc1 = wmma4(a, bB, acc1);
  }

  #pragma unroll
  for (int v = 0; v < 8; ++v) {
    const int m = v + 8 * hi;                  // C/D layout: M = v + 8*(lane>=16)
    sxp[m][lo]      = acc0[v];
    sxp[m][16 + lo] = acc1[v];
    xp_out[(size_t)(row0 + m) * D_IN + lo]      = acc0[v];
    xp_out[(size_t)(row0 + m) * D_IN + 16 + lo] = acc1[v];
  }
  __syncthreads();

  // ---- Stage 2: three GEMMs sharing A-fragments of xp (K=32, N=16)
  v8f aDt = {}, aB = {}, aC = {};
  #pragma unroll
  for (int k0 = 0; k0 < D_IN; k0 += 4) {
    const int k = k0 + 2 * hi;
    v2f a;  a.x  = sxp[lo][k];               a.y  = sxp[lo][k + 1];
    v2f bt; bt.x = Wdt[k * DT_RANK + lo];    bt.y = Wdt[(k + 1) * DT_RANK + lo];
    v2f bb; bb.x = WB[k * D_STATE + lo];     bb.y = WB[(k + 1) * D_STATE + lo];
    v2f bc; bc.x = WC[k * D_STATE + lo];     bc.y = WC[(k + 1) * D_STATE + lo];
    aDt = wmma4(a, bt, aDt);
    aB  = wmma4(a, bb, aB);
    aC  = wmma4(a, bc, aC);
  }
  #pragma unroll
  for (int v = 0; v < 8; ++v) {
    const int m = v + 8 * hi;
    sdt[m][lo] = aDt[v];
    sB[m][lo]  = aB[v];
    sC[m][lo]  = aC[v];
  }
  __syncthreads();

  // ---- dt[row] = softplus( (dtlin+bdt) . Wdtr )
  if (lane < 16) {
    float s = 0.f;
    #pragma unroll
    for (int n = 0; n < DT_RANK; ++n) s += (sdt[lane][n] + bdt[n]) * Wdtr[n];
    sdtv[lane] = fmaxf(s, 0.f) + log1pf(__expf(-fabsf(s))); // stable softplus
  }
  __syncthreads();

  // ---- emit Ad = exp(dt*A), Bd = dt*Bm, C
  #pragma unroll
  for (int i = 0; i < 8; ++i) {
    const int e = lane * 8 + i;
    const int m = e >> 4;
    const int n = e & 15;
    const float dtv = sdtv[m];
    const size_t g = (size_t)(row0 + m) * D_STATE + n;
    Ad_out[g] = __expf(dtv * Avec[n]);
    Bd_out[g] = dtv * sB[m][n];
    C_out[g]  = sC[m][n];
  }
}

// ---------------------------------------------------------------------------
// Kernel 2: selective scan. 1 block per batch, thread (d,n) owns h[d,n].
// Sequential over t; y_d = sum_n c_n*h reduced across the 16 n-lanes.
// global_prefetch_b8 streams the next 16-step window during the chain.
// ---------------------------------------------------------------------------
__global__ void __launch_bounds__(512)
scan_kernel(const float* __restrict__ xp, const float* __restrict__ Ad,
            const float* __restrict__ Bd, const float* __restrict__ Cm,
            float* __restrict__ y, float* __restrict__ h_rep)
{
  __shared__ float sh[D_IN * D_STATE];
  const int b   = blockIdx.x;
  const int tid = threadIdx.x;
  const int d   = tid >> 4;
  const int n   = tid & 15;

  const float* pa = Ad + (size_t)b * SEQ * D_STATE + n;
  const float* pb = Bd + (size_t)b * SEQ * D_STATE + n;
  const float* pc = Cm + (size_t)b * SEQ * D_STATE + n;
  const float* px = xp + (size_t)b * SEQ * D_IN + d;
  float*       py = y  + (size_t)b * SEQ * D_IN + d;

  float h = 0.f;
  for (int t = 0; t < SEQ; ++t) {
    if ((t & 15) == 0 && t + 16 < SEQ) {       // -> global_prefetch_b8
      __builtin_prefetch(pa + (size_t)(t + 16) * D_STATE, 0, 0);
      __builtin_prefetch(pb + (size_t)(t + 16) * D_STATE, 0, 0);
      __builtin_prefetch(pc + (size_t)(t + 16) * D_STATE, 0, 0);
      __builtin_prefetch(px + (size_t)(t + 16) * D_IN, 0, 0);
    }
    const float a  = pa[(size_t)t * D_STATE];
    const float bd = pb[(size_t)t * D_STATE];
    const float c  = pc[(size_t)t * D_STATE];
    const float xt = px[(size_t)t * D_IN];
    h = a * h + bd * xt;
    float yp = c * h;
    #pragma unroll
    for (int m = 1; m < 16; m <<= 1) yp += __shfl_xor(yp, m, 32);
    if (n == 0) py[(size_t)t * D_IN] = yp;
  }

  // h_rep[b,n] = mean over d of final h
  sh[tid] = h;
  __syncthreads();
  if (tid < D_STATE) {
    float s = 0.f;
    #pragma unroll
    for (int dd = 0; dd < D_IN; ++dd) s += sh[dd * D_STATE + tid];
    h_rep[b * D_STATE + tid] = s * (1.0f / D_IN);
  }
}

// ---------------------------------------------------------------------------
// Kernel 3: Sx[b,d] = sum_s x, Sxx[b,d] = sum_s x^2  (for closed-form loss)
// ---------------------------------------------------------------------------
__global__ void __launch_bounds__(256)
xstats_kernel(const float* __restrict__ x, float* __restrict__ Sx,
              float* __restrict__ Sxx)
{
  __shared__ float s1[256], s2[256];
  const int bd = blockIdx.x;                // b*32+d
  const int b = bd >> 5, d = bd & 31;
  const int tid = threadIdx.x;
  float a1 = 0.f, a2 = 0.f;
  for (int s = tid; s < SEQ; s += 256) {
    const float v = x[((size_t)b * SEQ + s) * D_IN + d];
    a1 += v; a2 += v * v;
  }
  s1[tid] = a1; s2[tid] = a2; __syncthreads();
  for (int off = 128; off > 0; off >>= 1) {
    if (tid < off) { s1[tid] += s1[tid + off]; s2[tid] += s2[tid + off]; }
    __syncthreads();
  }
  if (tid == 0) { Sx[bd] = s1[0]; Sxx[bd] = s2[0]; }
}

// ---------------------------------------------------------------------------
// Kernel 4: decode MLP + closed-form MSE:
//   sum_s (u-x)^2 = S*u^2 - 2*u*Sx + Sxx  -> mean over (b,s), sum over d
// ---------------------------------------------------------------------------
__global__ void __launch_bounds__(128)
mlp_loss_kernel(const float* __restrict__ h_rep,
                const float* __restrict__ Wd1, const float* __restrict__ bd1,
                const float* __restrict__ Wd2, const float* __restrict__ bd2,
                const float* __restrict__ Wd3, const float* __restrict__ bd3,
                const float* __restrict__ Sx, const float* __restrict__ Sxx,
                float* __restrict__ loss_out)
{
  __shared__ float z1[BATCH][HID];
  __shared__ float z2[BATCH][HID];
  __shared__ float red[D_IN];
  const int j = threadIdx.x;

  for (int b = 0; b < BATCH; ++b) {
    float s = bd1[j];
    #pragma unroll
    for (int n = 0; n < D_STATE; ++n) s += h_rep[b * D_STATE + n] * Wd1[n * HID + j];
    z1[b][j] = fmaxf(s, 0.f);
  }
  __syncthreads();
  for (int b = 0; b < BATCH; ++b) {
    float s = bd2[j];
    for (int k = 0; k < HID; ++k) s += z1[b][k] * Wd2[k * HID + j];
    z2[b][j] = fmaxf(s, 0.f);
  }
  __syncthreads();
  if (j < D_IN) {
    float acc = 0.f;
    for (int b = 0; b < BATCH; ++b) {
      float u = bd3[j];
      for (int k = 0; k < HID; ++k) u += z2[b][k] * Wd3[k * D_IN + j];
      acc += (float)SEQ * u * u - 2.f * u * Sx[b * D_IN + j] + Sxx[b * D_IN + j];
    }
    red[j] = acc;
  }
  __syncthreads();
  if (j == 0) {
    float t = 0.f;
    for (int d = 0; d < D_IN; ++d) t += red[d];
    loss_out[0] = t / (float)(BATCH * SEQ);
  }
}

// ---------------------------------------------------------------------------
// Kernel 5: out[b,s,d,:] = y[b,s,d]*Wout + bout -- 134 MB of float4 stores,
// wide grid so the write hits full HBM bandwidth (the true roofline floor).
// ---------------------------------------------------------------------------
__global__ void __launch_bounds__(256)
expand_kernel(const float* __restrict__ y, const float* __restrict__ Wout,
              const float* __restrict__ bout, float4* __restrict__ out)
{
  const size_t gid = (size_t)blockIdx.x * blockDim.x + threadIdx.x; // float4 idx
  const int j4 = (int)(gid & 7);          // 32 cols / 4
  const size_t rd = gid >> 3;             // (bs*32 + d)
  const float yv = y[rd];
  const float4 w  = ((const float4*)Wout)[j4];
  const float4 bo = ((const float4*)bout)[j4];
  float4 o;
  o.x = yv * w.x + bo.x; o.y = yv * w.y + bo.y;
  o.z = yv * w.z + bo.z; o.w = yv * w.w + bo.w;
  out[gid] = o;
}

// ---------------------------------------------------------------------------
extern "C" void kernel_launch(void* const* d_in, const int* in_sizes, int n_in,
                              void* d_out, int out_size, void* d_ws, size_t ws_size,
                              hipStream_t stream)
{
  const float* x    = (const float*)d_in[0];
  const float* Avec = (const float*)d_in[1];
  const float* Win  = (const float*)d_in[2];
  const float* bin_ = (const float*)d_in[3];
  const float* Wdt  = (const float*)d_in[4];
  const float* bdt  = (const float*)d_in[5];
  const float* Wdtr = (const float*)d_in[6];
  const float* WB   = (const float*)d_in[7];
  const float* WC   = (const float*)d_in[8];
  const float* Wout = (const float*)d_in[9];
  const float* bout = (const float*)d_in[10];
  const float* Wd1  = (const float*)d_in[11];
  const float* bd1  = (const float*)d_in[12];
  const float* Wd2  = (const float*)d_in[13];
  const float* bd2  = (const float*)d_in[14];
  const float* Wd3  = (const float*)d_in[15];
  const float* bd3  = (const float*)d_in[16];

  float* out = (float*)d_out;

  char* ws = (char*)d_ws;
  float* xp   = (float*)ws; ws += (size_t)ROWS * D_IN * sizeof(float);
  float* Ad   = (float*)ws; ws += (size_t)ROWS * D_STATE * sizeof(float);
  float* Bd   = (float*)ws; ws += (size_t)ROWS * D_STATE * sizeof(float);
  float* Cm   = (float*)ws; ws += (size_t)ROWS * D_STATE * sizeof(float);
  float* y    = (float*)ws; ws += (size_t)ROWS * D_IN * sizeof(float);
  float* hrep = (float*)ws; ws += BATCH * D_STATE * sizeof(float);
  float* Sx   = (float*)ws; ws += BATCH * D_IN * sizeof(float);
  float* Sxx  = (float*)ws; ws += BATCH * D_IN * sizeof(float);

  proj_kernel<<<ROWS / 16, 32, 0, stream>>>(x, Avec, Win, bin_, Wdt, bdt, Wdtr,
                                            WB, WC, xp, Ad, Bd, Cm);
  xstats_kernel<<<BATCH * D_IN, 256, 0, stream>>>(x, Sx, Sxx);
  scan_kernel<<<BATCH, 512, 0, stream>>>(xp, Ad, Bd, Cm, y, hrep);
  mlp_loss_kernel<<<1, 128, 0, stream>>>(hrep, Wd1, bd1, Wd2, bd2, Wd3, bd3,
                                         Sx, Sxx, out + OUT_MAIN);
  expand_kernel<<<(unsigned)((OUT_MAIN / 4) / 256), 256, 0, stream>>>(
      y, Wout, bout, (float4*)out);
}